// HeteroEmbedding_77902116815496
// MI455X (gfx1250) — compile-verified
//
#include <hip/hip_runtime.h>
#include <hip/hip_bf16.h>

// HeteroEmbedding gather: out[n] = tables[types[n], x[n]], padding_idx=0.
// N = 2,000,000 rows, EMBED = 128 floats (512 B) per row, VOCAB = 50,000,
// NUM_TYPES = 8. Pure bandwidth-bound gather: ~2.06 GB moved -> ~88 us floor
// at 23.3 TB/s HBM.
//
// Mapping: one wave32 per row; lane l moves bytes [16l,16l+16) of the row
// (32 lanes x 16 B = 512 B = one row). Each wave handles 4 rows to keep 4
// b128 gathers in flight. Index quads are fetched as single b128 loads.
// All hot-path address math is 32-bit (max offset 256M elements < 2^31);
// the only bounds check is one wave-uniform branch. Output stores are
// non-temporal so the 1 GB write-once stream doesn't evict the 204.8 MB
// table from the 192 MB L2.

typedef float v4f __attribute__((ext_vector_type(4)));
typedef int   v4i __attribute__((ext_vector_type(4)));

#define EMBED_F       128u
#define VOCAB_SZ      50000u
#define ROWS_PER_WAVE 4
#define WAVES_PER_BLK 8
#define BLOCK_SZ      (WAVES_PER_BLK * 32)
#define ROWS_PER_BLK  (WAVES_PER_BLK * ROWS_PER_WAVE)

__global__ __launch_bounds__(BLOCK_SZ)
void HeteroEmbedding_77902116815496_kernel(const int* __restrict__ x,
                                           const int* __restrict__ types,
                                           const float* __restrict__ tables,
                                           float* __restrict__ out,
                                           int n) {
    const unsigned lane = threadIdx.x & 31u;          // wave32 lane
    const unsigned wave = threadIdx.x >> 5;           // wave within block
    const unsigned rowBase =
        (blockIdx.x * (unsigned)WAVES_PER_BLK + wave) * (unsigned)ROWS_PER_WAVE;
    const unsigned laneOff = lane * 4u;               // float offset in row

    if (rowBase + ROWS_PER_WAVE <= (unsigned)n) {
        // ---- fast path: full tile, no per-row guards (wave-uniform) ----
        // Index quads as single b128 loads (rowBase is 16B-aligned).
        const v4i xiv = *(const v4i*)(x + rowBase);
        const v4i tiv = *(const v4i*)(types + rowBase);

        // Issue all 4 gathers before any store: 4 outstanding b128 loads.
        v4f v[ROWS_PER_WAVE];
#pragma unroll
        for (int r = 0; r < ROWS_PER_WAVE; ++r) {
            const unsigned xi  = (unsigned)xiv[r];
            const unsigned ti  = (unsigned)tiv[r];
            const unsigned off = (ti * VOCAB_SZ + xi) * EMBED_F + laneOff;
            const v4f t = *(const v4f*)(tables + (size_t)off);
            const v4f z = {0.f, 0.f, 0.f, 0.f};
            v[r] = (xiv[r] != 0) ? t : z;   // padding_idx=0 (row 0 is zero anyway)
        }

        const unsigned outBase = rowBase * EMBED_F + laneOff;
#pragma unroll
        for (int r = 0; r < ROWS_PER_WAVE; ++r) {
            v4f* dst = (v4f*)(out + (size_t)(outBase + (unsigned)r * EMBED_F));
            __builtin_nontemporal_store(v[r], dst);
        }
    } else {
        // ---- remainder path: per-row guarded (rare / never for N%32==0) ----
        for (int r = 0; r < ROWS_PER_WAVE; ++r) {
            const unsigned row = rowBase + (unsigned)r;
            if (row < (unsigned)n) {
                const int      xi  = x[row];
                const unsigned ti  = (unsigned)types[row];
                const unsigned off =
                    (ti * VOCAB_SZ + (unsigned)xi) * EMBED_F + laneOff;
                const v4f t = *(const v4f*)(tables + (size_t)off);
                const v4f z = {0.f, 0.f, 0.f, 0.f};
                const v4f v = (xi != 0) ? t : z;
                v4f* dst = (v4f*)(out + (size_t)(row * EMBED_F + laneOff));
                __builtin_nontemporal_store(v, dst);
            }
        }
    }
}

extern "C" void kernel_launch(void* const* d_in, const int* in_sizes, int n_in,
                              void* d_out, int out_size, void* d_ws, size_t ws_size,
                              hipStream_t stream) {
    const int*   x      = (const int*)d_in[0];     // [N]
    const int*   types  = (const int*)d_in[1];     // [N]
    const float* tables = (const float*)d_in[2];   // [NUM_TYPES, VOCAB, EMBED]
    float*       out    = (float*)d_out;           // [N, EMBED]

    const int n = in_sizes[0];
    const int grid = (n + ROWS_PER_BLK - 1) / ROWS_PER_BLK;

    HeteroEmbedding_77902116815496_kernel<<<grid, BLOCK_SZ, 0, stream>>>(
        x, types, tables, out, n);
}